// GraphInductiveLayer_36447092474026
// MI455X (gfx1250) — compile-verified
//
#include <hip/hip_runtime.h>

typedef __attribute__((ext_vector_type(2))) float v2f;
typedef __attribute__((ext_vector_type(4))) float v4f;
typedef __attribute__((ext_vector_type(8))) float v8f;

#define S_SAMP 25
#define F      128      // F_IN == F_OUT == 128
#define TILE_M 32       // nodes per block (two 16-row WMMA tiles)

// One block = 256 threads = 8 wave32, 32 nodes per block.
// Phase 1: stage 32*25 neighbor indices in LDS.
// Phase 2: wave w aggregates nodes {4w..4w+3}; each lane owns 4 consecutive
//          feature columns (float4) -> every gathered row is one fully
//          coalesced 512B wave read (x is 51 MB, L2-resident: 192 MB L2).
//          Row base is readfirstlane'd to SGPRs (index is wave-uniform).
// Phase 3: wave w computes output columns [16w, 16w+16) for BOTH 16-row
//          tiles, reusing each B fragment for two WMMAs (64x
//          v_wmma_f32_16x16x4_f32 per wave) -> halves W read traffic vs a
//          16-row block. A from LDS (ds_load_b64), B from global W.
__global__ __launch_bounds__(256)
void graphsage_fused_kernel(const float* __restrict__ x,
                            const int*   __restrict__ nidx,
                            const float* __restrict__ W,
                            const float* __restrict__ bias,
                            float*       __restrict__ out)
{
    __shared__ float agg_lds[TILE_M * F];       // 16 KB: mean-aggregated tile
    __shared__ int   idx_lds[TILE_M * S_SAMP];  // 3.2 KB: staged indices

    const int tid  = threadIdx.x;
    const int wave = tid >> 5;
    const int lane = tid & 31;
    const int row0 = blockIdx.x * TILE_M;       // first node of this tile

    // ---- Phase 1: stage neighbor indices --------------------------------
    for (int i = tid; i < TILE_M * S_SAMP; i += 256)
        idx_lds[i] = nidx[(size_t)row0 * S_SAMP + i];
    __syncthreads();

    // ---- Phase 2: gather + mean aggregation -----------------------------
    const int   c   = lane * 4;                 // 4 feature cols per lane
    const float inv = 1.0f / (float)(S_SAMP + 1);
    #pragma unroll
    for (int r = 0; r < 4; ++r) {
        const int nloc = wave * 4 + r;          // local node 0..31
        const int node = row0 + nloc;
        v4f acc = *(const v4f*)(x + (size_t)node * F + c);   // self feature
        #pragma unroll 5
        for (int s = 0; s < S_SAMP; ++s) {
            // wave-uniform index -> force into SGPR so the gather becomes
            // global_load_b128 vdst, voffset, s[base:base+1]
            const int gi = __builtin_amdgcn_readfirstlane(
                idx_lds[nloc * S_SAMP + s]);
            const float* row = x + (size_t)gi * F;           // SGPR base
            acc += *(const v4f*)(row + c);                   // 512B/wave
        }
        acc *= inv;
        *(v4f*)(agg_lds + nloc * F + c) = acc;               // b128 DS store
    }
    __syncthreads();

    // ---- Phase 3: two 16x16 WMMA tiles over K=128 -----------------------
    // A frag (16x4 f32): lanes 0-15 -> M=lane, K={k,k+1}; lanes 16-31 -> K={k+2,k+3}
    // B frag (4x16 f32): lane%16 -> N, lane-half selects K pair (mirror of A)
    const int c0 = wave * 16;                   // this wave's output col base
    const int nl = lane & 15;
    const int hi = lane >> 4;
    v8f accC0 = {};                             // rows row0 +  0..15
    v8f accC1 = {};                             // rows row0 + 16..31
    for (int k = 0; k < F; k += 4) {
        const int kb = k + 2 * hi;
        const v2f a0 = *(const v2f*)(agg_lds + nl * F + kb);        // ds b64
        const v2f a1 = *(const v2f*)(agg_lds + (nl + 16) * F + kb); // ds b64
        v2f b;
        b.x = W[(size_t)kb * F + c0 + nl];
        b.y = W[(size_t)(kb + 1) * F + c0 + nl];
        // (neg_a, A, neg_b, B, c_mod, C, reuse_a, reuse_b)
        accC0 = __builtin_amdgcn_wmma_f32_16x16x4_f32(
            false, a0, false, b, (short)0, accC0, false, false);
        accC1 = __builtin_amdgcn_wmma_f32_16x16x4_f32(
            false, a1, false, b, (short)0, accC1, false, false);
    }

    // ---- Epilogue: bias + store -----------------------------------------
    // C/D layout: VGPR v -> row M = v + 8*hi, col N = lane&15.
    const float bv = bias[c0 + nl];
    #pragma unroll
    for (int v = 0; v < 8; ++v) {
        const int m = v + 8 * hi;
        out[(size_t)(row0 + m) * F + c0 + nl]      = accC0[v] + bv;
        out[(size_t)(row0 + 16 + m) * F + c0 + nl] = accC1[v] + bv;
    }
}

extern "C" void kernel_launch(void* const* d_in, const int* in_sizes, int n_in,
                              void* d_out, int out_size, void* d_ws, size_t ws_size,
                              hipStream_t stream) {
    (void)n_in; (void)out_size; (void)d_ws; (void)ws_size;
    const float* x    = (const float*)d_in[0];   // [N, 128] f32
    const int*   nidx = (const int*)  d_in[1];   // [N, 25] int
    const float* W    = (const float*)d_in[2];   // [128, 128] f32
    const float* b    = (const float*)d_in[3];   // [128] f32
    float*       out  = (float*)d_out;           // [N, 128] f32

    const int n = in_sizes[0] / F;               // 100000 (divisible by 32)
    graphsage_fused_kernel<<<dim3(n / TILE_M), dim3(256), 0, stream>>>(
        x, nidx, W, b, out);
}